// GlobalSimilarityLearning_24077586661601
// MI455X (gfx1250) — compile-verified
//
#include <hip/hip_runtime.h>
#include <hip/hip_bf16.h>

typedef __attribute__((ext_vector_type(2))) float v2f;
typedef __attribute__((ext_vector_type(8))) float v8f;
typedef __attribute__((address_space(3))) float lds_f;

#define BSZ  1024
#define LSEQ 200
#define DMOD 128
#define KBLK 32        // staged K-depth per LDS tile
#define BSTRIDE 72     // padded floats per staged row: 288B rows (16B aligned),
                       // rows k and k+2 land in disjoint bank ranges -> conflict-free

static __device__ __forceinline__ unsigned lds_offset(const float* p) {
  // exact LDS byte offset via addrspace(3) ptrtoint
  return (unsigned)(unsigned long long)(const lds_f*)p;
}

// ---------------------------------------------------------------------------
// K1: g0[b,:] = sum_l item_embedding[items[b,l], :]   (HBM-bound gather)
// ---------------------------------------------------------------------------
__global__ __launch_bounds__(128) void gather_sum_kernel(
    const float* __restrict__ item_emb, const long long* __restrict__ items,
    float* __restrict__ g0) {
  const int b = blockIdx.x;
  const int t = threadIdx.x;
  const long long* row = items + (size_t)b * LSEQ;
  float acc = 0.f;
  long long idx = row[0];
  for (int l = 0; l < LSEQ; ++l) {
    long long next = (l + 1 < LSEQ) ? row[l + 1] : idx;
    __builtin_prefetch(item_emb + (size_t)next * DMOD + t, 0, 0);  // global_prefetch_b8
    acc += item_emb[(size_t)idx * DMOD + t];
    idx = next;
  }
  g0[b * DMOD + t] = acc;
}

// ---------------------------------------------------------------------------
// f32 WMMA GEMM with async-to-LDS double-buffered B staging.
// Block: 256 threads = 8 waves. Block tile: 128(M) x 64(N).
// Each wave: 16x64 output (4 accumulators), A-fragment reused across 4 WMMAs.
// grid = (N/64, M/128). Requires M%128==0, N%64==0, K%KBLK==0.
// ---------------------------------------------------------------------------
__global__ __launch_bounds__(256) void wmma_gemm_f32_async_kernel(
    const float* __restrict__ A, const float* __restrict__ B,
    float* __restrict__ C, int M, int N, int K) {
  __shared__ float btile[2][KBLK * BSTRIDE];

  const int tid  = threadIdx.x;
  const int wave = tid >> 5;
  const int lane = tid & 31;
  const int half = lane >> 4;      // lanes 16-31: +2 K slots / +8 M rows
  const int l15  = lane & 15;
  const int tm  = blockIdx.y * 128 + wave * 16;
  const int tn0 = blockIdx.x * 64;

  const float* Arow = A + (size_t)(tm + l15) * K;

  // Each thread stages 2 x b128 chunks per KBLKx64 tile (32*64 floats).
  const int c0 = tid, c1 = tid + 256;
  const int r0 = c0 >> 4, q0 = (c0 & 15) << 2;
  const int r1 = c1 >> 4, q1 = (c1 & 15) << 2;

  v8f acc[4] = {v8f{}, v8f{}, v8f{}, v8f{}};

  const int nk = K / KBLK;
  int buf = 0;

  // prologue: stage tile 0 asynchronously
  {
    const float* s0 = B + (size_t)r0 * N + tn0 + q0;
    const float* s1 = B + (size_t)r1 * N + tn0 + q1;
    unsigned d0 = lds_offset(&btile[0][r0 * BSTRIDE + q0]);
    unsigned d1 = lds_offset(&btile[0][r1 * BSTRIDE + q1]);
    asm volatile("global_load_async_to_lds_b128 %0, %1, off" :: "v"(d0), "v"(s0) : "memory");
    asm volatile("global_load_async_to_lds_b128 %0, %1, off" :: "v"(d1), "v"(s1) : "memory");
  }

  for (int i = 0; i < nk; ++i) {
    const int k0 = i * KBLK;
    if (i + 1 < nk) {
      // issue next tile into the other buffer, then wait for the current one
      const float* s0 = B + (size_t)(k0 + KBLK + r0) * N + tn0 + q0;
      const float* s1 = B + (size_t)(k0 + KBLK + r1) * N + tn0 + q1;
      unsigned d0 = lds_offset(&btile[buf ^ 1][r0 * BSTRIDE + q0]);
      unsigned d1 = lds_offset(&btile[buf ^ 1][r1 * BSTRIDE + q1]);
      asm volatile("global_load_async_to_lds_b128 %0, %1, off" :: "v"(d0), "v"(s0) : "memory");
      asm volatile("global_load_async_to_lds_b128 %0, %1, off" :: "v"(d1), "v"(s1) : "memory");
      asm volatile("s_wait_asynccnt 0x2" ::: "memory");   // current tile resident
    } else {
      asm volatile("s_wait_asynccnt 0x0" ::: "memory");
    }
    __syncthreads();

    const float* bt = &btile[buf][0];
#pragma unroll
    for (int kk = 0; kk < KBLK; kk += 4) {
      const int ka = kk + half * 2;
      v2f af;
      af.x = Arow[k0 + ka];
      af.y = Arow[k0 + ka + 1];
#pragma unroll
      for (int j = 0; j < 4; ++j) {
        const int n = j * 16 + l15;
        v2f bf;
        bf.x = bt[ka * BSTRIDE + n];
        bf.y = bt[(ka + 1) * BSTRIDE + n];
        acc[j] = __builtin_amdgcn_wmma_f32_16x16x4_f32(false, af, false, bf,
                                                       (short)0, acc[j], false, false);
      }
    }
    __syncthreads();   // protect buffer before it is overwritten next iteration
    buf ^= 1;
  }

#pragma unroll
  for (int j = 0; j < 4; ++j) {
#pragma unroll
    for (int r = 0; r < 8; ++r) {
      const int m = tm + r + half * 8;
      C[(size_t)m * N + tn0 + j * 16 + l15] = acc[j][r];
    }
  }
}

// ---------------------------------------------------------------------------
// H = relu(G@W1 + Q@W2 + bias), G,Q,H:[1024,128], W1,W2:[128,128]
// Both GEMMs accumulate into the same WMMA accumulator. grid=(8,16), block=128.
// ---------------------------------------------------------------------------
__global__ __launch_bounds__(128) void attn_h_kernel(
    const float* __restrict__ G, const float* __restrict__ Q,
    const float* __restrict__ W1, const float* __restrict__ W2,
    const float* __restrict__ bias, float* __restrict__ H) {
  const int wave = threadIdx.x >> 5;
  const int lane = threadIdx.x & 31;
  const int half = lane >> 4;
  const int l15  = lane & 15;
  const int tm = blockIdx.y * 64 + wave * 16;
  const int tn = blockIdx.x * 16;
  const int K = DMOD, N = DMOD;

  v8f c = {};
  const float* Gr = G + (size_t)(tm + l15) * K;
#pragma unroll 4
  for (int k = 0; k < K; k += 4) {
    const int ka = k + half * 2;
    v2f af, bf;
    af.x = Gr[ka];            af.y = Gr[ka + 1];
    bf.x = W1[ka * N + tn + l15];
    bf.y = W1[(ka + 1) * N + tn + l15];
    c = __builtin_amdgcn_wmma_f32_16x16x4_f32(false, af, false, bf,
                                              (short)0, c, false, false);
  }
  const float* Qr = Q + (size_t)(tm + l15) * K;
#pragma unroll 4
  for (int k = 0; k < K; k += 4) {
    const int ka = k + half * 2;
    v2f af, bf;
    af.x = Qr[ka];            af.y = Qr[ka + 1];
    bf.x = W2[ka * N + tn + l15];
    bf.y = W2[(ka + 1) * N + tn + l15];
    c = __builtin_amdgcn_wmma_f32_16x16x4_f32(false, af, false, bf,
                                              (short)0, c, false, false);
  }
  const float bcol = bias[tn + l15];
#pragma unroll
  for (int r = 0; r < 8; ++r) {
    const int m = tm + r + half * 8;
    const float v = c[r] + bcol;
    H[(size_t)m * N + tn + l15] = v > 0.f ? v : 0.f;
  }
}

// ---------------------------------------------------------------------------
// score[b] = dot(H[b,:], w0);  alpha[b] = sigmoid(dot(Q[b,:],aw)+ab)+1
// ---------------------------------------------------------------------------
__global__ __launch_bounds__(128) void score_alpha_kernel(
    const float* __restrict__ H, const float* __restrict__ Q,
    const float* __restrict__ w0, const float* __restrict__ aw,
    const float* __restrict__ ab, float* __restrict__ score,
    float* __restrict__ alpha) {
  __shared__ float s1[DMOD];
  __shared__ float s2[DMOD];
  const int b = blockIdx.x;
  const int t = threadIdx.x;
  s1[t] = H[b * DMOD + t] * w0[t];
  s2[t] = Q[b * DMOD + t] * aw[t];
  __syncthreads();
  for (int off = 64; off > 0; off >>= 1) {
    if (t < off) { s1[t] += s1[t + off]; s2[t] += s2[t + off]; }
    __syncthreads();
  }
  if (t == 0) {
    score[b] = s1[0];
    float a = 1.f / (1.f + __expf(-(s2[0] + ab[0]))) + 1.f;
    if (a == 1.f) a = 1.00001f;
    alpha[b] = a;
  }
}

// ---------------------------------------------------------------------------
// scalar entmax bisection (scores constant over L -> one scalar per b),
// then c = (sum_l p)*g, selu, L2-normalize.
// ---------------------------------------------------------------------------
__global__ __launch_bounds__(128) void entmax_out_kernel(
    const float* __restrict__ G, const float* __restrict__ score,
    const float* __restrict__ alpha, float* __restrict__ out) {
  const int b = blockIdx.x;
  const int t = threadIdx.x;
  const float a   = alpha[b];
  const float s   = score[b];
  const float am1 = a - 1.f;
  const float inv = 1.f / am1;
  const float Xs  = s * am1;
  const float L   = (float)LSEQ;

  float tau_lo = Xs - 1.f;
  const float tau_hi = Xs - __powf(1.f / L, am1);
  const float f_lo = L - 1.f;
  float dm = tau_hi - tau_lo;
  float p = 1.f;
  for (int it = 0; it < 50; ++it) {
    dm *= 0.5f;
    const float tau_m = tau_lo + dm;
    const float z = Xs - tau_m;
    const float pm = (z > 0.f) ? __powf(z, inv) : 0.f;
    const float fm = L * pm - 1.f;
    if (fm * f_lo >= 0.f) tau_lo = tau_m;
    p = pm;
  }
  const float psum = L * p;
  const float factor = (psum > 0.f) ? (L * p / psum) : 1.f;  // == 1 analytically

  const float cj = factor * G[b * DMOD + t];
  const float sc = 1.0507009873554805f, al = 1.6732632423543772f;
  const float se = sc * (cj > 0.f ? cj : al * (__expf(cj) - 1.f));

  __shared__ float ss[DMOD];
  ss[t] = se * se;
  __syncthreads();
  for (int off = 64; off > 0; off >>= 1) {
    if (t < off) ss[t] += ss[t + off];
    __syncthreads();
  }
  out[b * DMOD + t] = se * rsqrtf(ss[0]);
}

// ---------------------------------------------------------------------------
extern "C" void kernel_launch(void* const* d_in, const int* in_sizes, int n_in,
                              void* d_out, int out_size, void* d_ws, size_t ws_size,
                              hipStream_t stream) {
  const float*     item_emb = (const float*)d_in[0];
  const long long* items    = (const long long*)d_in[1];
  const float*     Amat     = (const float*)d_in[2];
  const float*     Dmat     = (const float*)d_in[3];
  const float*     Q        = (const float*)d_in[4];
  const float*     w0       = (const float*)d_in[5];
  const float*     W1       = (const float*)d_in[6];
  const float*     W2       = (const float*)d_in[7];
  const float*     bias     = (const float*)d_in[8];
  const float*     aw       = (const float*)d_in[9];
  const float*     ab       = (const float*)d_in[10];
  float* out = (float*)d_out;

  float* ws   = (float*)d_ws;
  float* g0   = ws;                    // [1024,128]
  float* DA   = g0 + BSZ * DMOD;       // [1024,1024]
  float* g1   = DA + BSZ * BSZ;        // [1024,128]
  float* H    = g1 + BSZ * DMOD;       // [1024,128]
  float* scr  = H  + BSZ * DMOD;       // [1024]
  float* alph = scr + BSZ;             // [1024]

  // 1) gather + sum over sequence
  gather_sum_kernel<<<BSZ, 128, 0, stream>>>(item_emb, items, g0);
  // 2) DA = D @ A  (1024^3, async-LDS staged WMMA)
  wmma_gemm_f32_async_kernel<<<dim3(BSZ / 64, BSZ / 128), 256, 0, stream>>>(
      Dmat, Amat, DA, BSZ, BSZ, BSZ);
  // 3) g = DA @ g0  (1024x128x1024)
  wmma_gemm_f32_async_kernel<<<dim3(DMOD / 64, BSZ / 128), 256, 0, stream>>>(
      DA, g0, g1, BSZ, DMOD, BSZ);
  // 4) H = relu(g@W1 + q@W2 + bias)
  attn_h_kernel<<<dim3(DMOD / 16, BSZ / 64), 128, 0, stream>>>(
      g1, Q, W1, W2, bias, H);
  // 5) scores + alpha
  score_alpha_kernel<<<BSZ, 128, 0, stream>>>(H, Q, w0, aw, ab, scr, alph);
  // 6) entmax (collapsed over constant L) + c + selu + normalize
  entmax_out_kernel<<<BSZ, 128, 0, stream>>>(g1, scr, alph, out);
}